// LuongAttnDecoderRNN_34703335751727
// MI455X (gfx1250) — compile-verified
//
#include <hip/hip_runtime.h>
#include <hip/hip_bf16.h>

typedef __attribute__((ext_vector_type(2))) float v2f;
typedef __attribute__((ext_vector_type(8))) float v8f;
typedef int v4i __attribute__((vector_size(16)));   // matches builtin param type

#define BK  64
#define BKP 68   // +4 floats pad: row stride 272B -> conflict-free 16-row column reads

#if __has_builtin(__builtin_amdgcn_global_load_async_to_lds_b128)
#define HAS_ASYNC_LDS 1
#else
#define HAS_ASYNC_LDS 0
#endif

#if HAS_ASYNC_LDS
typedef __attribute__((address_space(1))) v4i* gv4i_p;
typedef __attribute__((address_space(3))) v4i* lv4i_p;

__device__ __forceinline__ void async_copy16(const float* g, float* l) {
    // global_load_async_to_lds_b128: per-lane 16B global -> LDS, ASYNCcnt tracked
    __builtin_amdgcn_global_load_async_to_lds_b128(
        (gv4i_p)(uintptr_t)g,
        (lv4i_p)(uint32_t)(uintptr_t)l,
        0, 0);
}
#endif

__device__ __forceinline__ void wait_async_lds() {
#if HAS_ASYNC_LDS
#if __has_builtin(__builtin_amdgcn_s_wait_asynccnt)
    __builtin_amdgcn_s_wait_asynccnt(0);
#else
    asm volatile("s_wait_asynccnt 0" ::: "memory");
#endif
#endif
}

// ---------------------------------------------------------------------------
// C[64 x N] = A[64 x K] @ W[N x K]^T + bias, optional tanh epilogue.
// 4 waves / block, wave w owns rows 16w..16w+15; block owns 16 output columns.
// Double-buffered LDS pipeline: async-copy K-chunk i+1 while 16x
// V_WMMA_F32_16X16X4_F32 consume chunk i; one s_wait_asynccnt + barrier per chunk.
// ---------------------------------------------------------------------------
template <int ACT>
__global__ __launch_bounds__(128) void gemm_xt_wmma(
    const float* __restrict__ A, const float* __restrict__ W,
    const float* __restrict__ bias, float* __restrict__ C,
    int N, int K)
{
    __shared__ float As[2][64][BKP];
    __shared__ float Ws[2][16][BKP];

    const int tid   = threadIdx.x;
    const int lane  = tid & 31;
    const int wave  = tid >> 5;       // m-tile
    const int half  = lane >> 4;      // K sub-pair selector (ISA f32 A/B layout)
    const int l16   = lane & 15;
    const int n0    = blockIdx.x * 16;
    const int mbase = wave * 16;

    v8f acc = {};

    auto stage = [&](int buf, int k0) {
#if HAS_ASYNC_LDS
#pragma unroll
        for (int it = 0; it < 8; ++it) {             // A: 64 rows x 16 float4
            int idx = tid + it * 128;
            int row = idx >> 4, c4 = idx & 15;
            async_copy16(A + (size_t)row * K + k0 + c4 * 4, &As[buf][row][c4 * 4]);
        }
#pragma unroll
        for (int it = 0; it < 2; ++it) {             // W tile: 16 rows x 16 float4
            int idx = tid + it * 128;
            int row = idx >> 4, c4 = idx & 15;
            async_copy16(W + (size_t)(n0 + row) * K + k0 + c4 * 4, &Ws[buf][row][c4 * 4]);
        }
#else
        float4 ta[8], tw[2];
#pragma unroll
        for (int it = 0; it < 8; ++it) {             // issue all loads first
            int idx = tid + it * 128;
            int row = idx >> 4, c4 = idx & 15;
            ta[it] = *reinterpret_cast<const float4*>(A + (size_t)row * K + k0 + c4 * 4);
        }
#pragma unroll
        for (int it = 0; it < 2; ++it) {
            int idx = tid + it * 128;
            int row = idx >> 4, c4 = idx & 15;
            tw[it] = *reinterpret_cast<const float4*>(W + (size_t)(n0 + row) * K + k0 + c4 * 4);
        }
#pragma unroll
        for (int it = 0; it < 8; ++it) {             // then all LDS stores
            int idx = tid + it * 128;
            int row = idx >> 4, c4 = idx & 15;
            *reinterpret_cast<float4*>(&As[buf][row][c4 * 4]) = ta[it];
        }
#pragma unroll
        for (int it = 0; it < 2; ++it) {
            int idx = tid + it * 128;
            int row = idx >> 4, c4 = idx & 15;
            *reinterpret_cast<float4*>(&Ws[buf][row][c4 * 4]) = tw[it];
        }
#endif
    };

    stage(0, 0);
    wait_async_lds();
    __syncthreads();

    int buf = 0;
    for (int k0 = 0; k0 < K; k0 += BK) {
        if (k0 + BK < K) stage(buf ^ 1, k0 + BK);    // prefetch next chunk (async)

#pragma unroll
        for (int kk = 0; kk < BK; kk += 4) {
            const int kc = kk + half * 2;
            // A 16x4 f32 frag: lanes 0-15 hold K={0,1}, lanes 16-31 K={2,3}
            float2 av = *reinterpret_cast<const float2*>(&As[buf][mbase + l16][kc]);
            // B 4x16 f32 frag: lane = column n, same K split
            float2 bv = *reinterpret_cast<const float2*>(&Ws[buf][l16][kc]);
            v2f a; a.x = av.x; a.y = av.y;
            v2f b; b.x = bv.x; b.y = bv.y;
            acc = __builtin_amdgcn_wmma_f32_16x16x4_f32(
                false, a, false, b, (short)0, acc, false, false);
        }

        wait_async_lds();        // next buffer fully landed in LDS
        __syncthreads();         // all waves done reading current buffer
        buf ^= 1;
    }

    // C/D layout: VGPR j -> row j (lanes 0-15) / row j+8 (lanes 16-31), col = l16
    const float bvv = bias[n0 + l16];
#pragma unroll
    for (int j = 0; j < 8; ++j) {
        float v = acc[j] + bvv;
        if (ACT == 1) v = tanhf(v);
        C[(size_t)(mbase + half * 8 + j) * N + n0 + l16] = v;
    }
}

// ---------------------------------------------------------------------------
__global__ __launch_bounds__(256) void gather_embed_kernel(
    const int* __restrict__ idx, const float* __restrict__ emb,
    float* __restrict__ x)
{
    int i = blockIdx.x * 256 + threadIdx.x;   // 64*1024
    int b = i >> 10, h = i & 1023;
    x[i] = emb[(size_t)idx[b] * 1024 + h];
}

__global__ __launch_bounds__(256) void gru_gates_kernel(
    const float* __restrict__ gi, const float* __restrict__ gh,
    const float* __restrict__ hprev,
    float* __restrict__ concat_in, float* __restrict__ hidden_out)
{
    int i = blockIdx.x * 256 + threadIdx.x;   // 64*1024
    int b = i >> 10, j = i & 1023;
    const float* gib = gi + (size_t)b * 3072;
    const float* ghb = gh + (size_t)b * 3072;
    float r = 1.f / (1.f + __expf(-(gib[j]        + ghb[j])));
    float z = 1.f / (1.f + __expf(-(gib[1024 + j] + ghb[1024 + j])));
    float n = tanhf(gib[2048 + j] + r * ghb[2048 + j]);
    float hp = hprev[i];
    float hn = (1.f - z) * n + z * hp;
    concat_in[(size_t)b * 2048 + j] = hn;     // left half of [h_new, context]
    hidden_out[i] = hn;
}

// ---------------------------------------------------------------------------
// Fused online-softmax attention partials. Block = (b, 128-s chunk).
// Per 32-s sub-chunk: phase A computes the 32 dot-products (8 waves x 4 rows,
// contiguous 4KB encoder rows); phase B immediately re-reads those same rows
// (L2-hot: 128KB just touched) with flash-style rescaled accumulators.
// Encoder tensor crosses HBM exactly once.
// ---------------------------------------------------------------------------
__global__ __launch_bounds__(256) void attn_partial_kernel(
    const float* __restrict__ hnew /* row stride 2048 */,
    const float* __restrict__ enc,
    float* __restrict__ scores,     // B x S raw scores (for exact attn output)
    float* __restrict__ partials,   // (B*16) x 1024 unnormalized contexts
    float* __restrict__ ml)         // (B*16) x {m, l}
{
    __shared__ float sc[32];
    const int b     = blockIdx.x >> 4;
    const int chunk = blockIdx.x & 15;
    const int tid   = threadIdx.x;
    const int wave  = tid >> 5, lane = tid & 31;
    const int s0    = chunk * 128;
    const float* h  = hnew + (size_t)b * 2048;

    float m = -3.0e38f, l = 0.f;
    float a0 = 0.f, a1 = 0.f, a2 = 0.f, a3 = 0.f;

    for (int sub = 0; sub < 4; ++sub) {
        const int sb = s0 + sub * 32;
        // ---- phase A: scores for 32 rows ----
#pragma unroll
        for (int i = 0; i < 4; ++i) {
            const int sl = wave + 8 * i;                   // 0..31
            const float* e = enc + ((size_t)(sb + sl) * 64 + b) * 1024;
            float sum = 0.f;
#pragma unroll
            for (int q = 0; q < 8; ++q) {
                int c = (q * 32 + lane) * 4;
                float4 ev = *reinterpret_cast<const float4*>(e + c);
                float4 hv = *reinterpret_cast<const float4*>(h + c);
                sum += ev.x * hv.x + ev.y * hv.y + ev.z * hv.z + ev.w * hv.w;
            }
#pragma unroll
            for (int off = 16; off > 0; off >>= 1) sum += __shfl_xor(sum, off, 32);
            if (lane == 0) {
                sc[sl] = sum;
                scores[(size_t)b * 2048 + sb + sl] = sum;
            }
        }
        __syncthreads();

        // ---- phase B: online-softmax accumulate over the same 32 rows ----
        float sm = sc[0];
#pragma unroll
        for (int i = 1; i < 32; ++i) sm = fmaxf(sm, sc[i]);
        const float nm = fmaxf(m, sm);
        const float resc = __expf(m - nm);
        a0 *= resc; a1 *= resc; a2 *= resc; a3 *= resc; l *= resc;
        for (int i = 0; i < 32; ++i) {
            const float w = __expf(sc[i] - nm);
            const float* e = enc + ((size_t)(sb + i) * 64 + b) * 1024;  // L2 hit
            l  += w;
            a0 += w * e[tid];
            a1 += w * e[tid + 256];
            a2 += w * e[tid + 512];
            a3 += w * e[tid + 768];
        }
        m = nm;
        __syncthreads();   // protect sc[] before next sub-chunk overwrites it
    }

    float* p = partials + (size_t)blockIdx.x * 1024;
    p[tid] = a0; p[tid + 256] = a1; p[tid + 512] = a2; p[tid + 768] = a3;
    if (tid == 0) { ml[blockIdx.x * 2] = m; ml[blockIdx.x * 2 + 1] = l; }
}

// exact softmax weights from raw scores + per-chunk (m,l) stats
__global__ __launch_bounds__(256) void attn_write_kernel(
    const float* __restrict__ scores, const float* __restrict__ ml,
    float* __restrict__ attn_out)
{
    int i = blockIdx.x * 256 + threadIdx.x;   // 64*2048
    int b = i >> 11;
    float gm = -3.0e38f;
#pragma unroll
    for (int c = 0; c < 16; ++c) gm = fmaxf(gm, ml[(b * 16 + c) * 2]);
    float tot = 0.f;
#pragma unroll
    for (int c = 0; c < 16; ++c)
        tot += ml[(b * 16 + c) * 2 + 1] * __expf(ml[(b * 16 + c) * 2] - gm);
    attn_out[i] = __expf(scores[i] - gm) / tot;
}

// combine 16 rescaled chunk contexts -> normalized context vector
__global__ __launch_bounds__(256) void context_combine_kernel(
    const float* __restrict__ partials, const float* __restrict__ ml,
    float* __restrict__ concat_in)
{
    int i = blockIdx.x * 256 + threadIdx.x;   // 64*1024
    int b = i >> 10, hh = i & 1023;
    float gm = -3.0e38f;
#pragma unroll
    for (int c = 0; c < 16; ++c) gm = fmaxf(gm, ml[(b * 16 + c) * 2]);
    float tot = 0.f;
#pragma unroll
    for (int c = 0; c < 16; ++c)
        tot += ml[(b * 16 + c) * 2 + 1] * __expf(ml[(b * 16 + c) * 2] - gm);
    float s = 0.f;
#pragma unroll
    for (int c = 0; c < 16; ++c)
        s += __expf(ml[(b * 16 + c) * 2] - gm) * partials[(size_t)(b * 16 + c) * 1024 + hh];
    concat_in[(size_t)b * 2048 + 1024 + hh] = s / tot;   // right half of [h_new, context]
}

// ---------------------------------------------------------------------------
extern "C" void kernel_launch(void* const* d_in, const int* in_sizes, int n_in,
                              void* d_out, int out_size, void* d_ws, size_t ws_size,
                              hipStream_t stream) {
    const int*   input_seq   = (const int*)  d_in[0];
    const float* last_hidden = (const float*)d_in[1];
    const float* enc         = (const float*)d_in[2];
    const float* embedding   = (const float*)d_in[3];
    const float* w_ih        = (const float*)d_in[4];
    const float* w_hh        = (const float*)d_in[5];
    const float* b_ih        = (const float*)d_in[6];
    const float* b_hh        = (const float*)d_in[7];
    const float* concat_w    = (const float*)d_in[8];
    const float* concat_b    = (const float*)d_in[9];
    const float* out_w       = (const float*)d_in[10];
    const float* out_b       = (const float*)d_in[11];

    float* out        = (float*)d_out;
    float* out_output = out;                   // 64 x 50000
    float* out_hidden = out + 3200000;         // 1 x 64 x 1024
    float* out_attn   = out + 3265536;         // 64 x 1 x 2048

    float* ws        = (float*)d_ws;           // ~7.4 MB used
    float* x         = ws;                     // 64 x 1024
    float* gi        = ws + 65536;             // 64 x 3072
    float* gh        = ws + 262144;            // 64 x 3072
    float* concat_in = ws + 458752;            // 64 x 2048  [h_new | context]
    float* concat_o  = ws + 589824;            // 64 x 1024
    float* scores    = ws + 655360;            // 64 x 2048 raw scores
    float* partials  = ws + 786432;            // 64 x 16 x 1024
    float* ml        = ws + 1835008;           // 64 x 16 x 2 (m, l)

    gather_embed_kernel   <<<256,  256, 0, stream>>>(input_seq, embedding, x);
    gemm_xt_wmma<0>       <<<192,  128, 0, stream>>>(x,           w_ih,     b_ih,     gi,         3072, 1024);
    gemm_xt_wmma<0>       <<<192,  128, 0, stream>>>(last_hidden, w_hh,     b_hh,     gh,         3072, 1024);
    gru_gates_kernel      <<<256,  256, 0, stream>>>(gi, gh, last_hidden, concat_in, out_hidden);
    attn_partial_kernel   <<<1024, 256, 0, stream>>>(concat_in, enc, scores, partials, ml);
    attn_write_kernel     <<<512,  256, 0, stream>>>(scores, ml, out_attn);
    context_combine_kernel<<<256,  256, 0, stream>>>(partials, ml, concat_in);
    gemm_xt_wmma<1>       <<<64,   128, 0, stream>>>(concat_in,   concat_w, concat_b, concat_o,   1024, 2048);
    gemm_xt_wmma<0>       <<<3125, 128, 0, stream>>>(concat_o,    out_w,    out_b,    out_output, 50000, 1024);
}